// HeteroSGC_53549652247155
// MI455X (gfx1250) — compile-verified
//
#include <hip/hip_runtime.h>
#include <math.h>

#define NA  30000
#define NP  60000
#define FA  128
#define FP  256
#define HID 128
#define OUT 40

typedef float v2f __attribute__((ext_vector_type(2)));
typedef float v8f __attribute__((ext_vector_type(8)));

__device__ __forceinline__ void atomAddF(float* p, float v) {
  // hardware global_atomic_add_f32 (no CAS loop) -- only used for degree counts
  unsafeAtomicAdd(p, v);
}

// ---------------------------------------------------------------------------
// FP32 WMMA GEMM: C[M,N] = A[M,K] @ B[K,N] + bias (optional relu).
// One wave computes one 16x16 tile via V_WMMA_F32_16X16X4_F32, K in steps of 4.
// A operand layout (ISA 7.12.2): lanes 0-15 hold row M=lane, K={k0,k0+1};
// lanes 16-31 hold K={k0+2,k0+3}. B mirrored (rows striped across lanes).
// C/D: VGPR r -> M=r (lanes 0-15), M=r+8 (lanes 16-31), N = lane&15.
// ---------------------------------------------------------------------------
template <int RELU>
__global__ __launch_bounds__(256) void gemm_wmma(
    const float* __restrict__ A, const float* __restrict__ B,
    const float* __restrict__ bias, float* __restrict__ C,
    int M, int N, int K) {
  const int lane = threadIdx.x & 31;
  const int wave = threadIdx.x >> 5;
  const int tile = blockIdx.x * 8 + wave;
  const int tilesN = N >> 4;
  const int totalTiles = (M >> 4) * tilesN;
  if (tile >= totalTiles) return;
  const int tm = tile / tilesN;
  const int tn = tile - tm * tilesN;
  const int half = lane >> 4;     // 0: K pair {0,1}; 1: K pair {2,3}
  const int l15  = lane & 15;
  const int arow = tm * 16 + l15;
  const int bcol = tn * 16 + l15;
  const float* Ap = A + (size_t)arow * K;
  const float* Bp = B + bcol;
  v8f acc = {};
  for (int k0 = 0; k0 < K; k0 += 4) {
    const int kk = k0 + (half << 1);
    v2f a = { Ap[kk], Ap[kk + 1] };
    v2f b = { Bp[(size_t)kk * N], Bp[(size_t)(kk + 1) * N] };
    acc = __builtin_amdgcn_wmma_f32_16x16x4_f32(
        /*neg_a=*/false, a, /*neg_b=*/false, b,
        /*c_mod=*/(short)0, acc, /*reuse_a=*/false, /*reuse_b=*/false);
  }
  const float bv = bias[bcol];
#pragma unroll
  for (int r = 0; r < 8; ++r) {
    float v = acc[r] + bv;
    if (RELU) v = fmaxf(v, 0.0f);
    C[(size_t)(tm * 16 + half * 8 + r) * N + bcol] = v;
  }
}

// ---------------------------------------------------------------------------
// Degree counting (per edge type): deg_s[src]++ ; deg_d[dst]++  (fp32 counts)
// ---------------------------------------------------------------------------
__global__ void degree_kernel(const int* __restrict__ src, const int* __restrict__ dst,
                              float* __restrict__ deg_s, float* __restrict__ deg_d, int n) {
  int i = blockIdx.x * blockDim.x + threadIdx.x;
  if (i < n) {
    atomAddF(&deg_s[src[i]], 1.0f);
    atomAddF(&deg_d[dst[i]], 1.0f);
  }
}

__global__ void rsqrt_clamp_kernel(float* __restrict__ d, int n) {
  int i = blockIdx.x * blockDim.x + threadIdx.x;
  if (i < n) d[i] = rsqrtf(fmaxf(d[i], 1.0f));
}

// ---------------------------------------------------------------------------
// Exclusive scan over fp32 integer-valued degrees -> CSR row offsets.
// Single workgroup of 1024 threads (32 waves), chunked; n <= ~60K so trivial.
// off must have n+1 entries; off[n] = total.
// ---------------------------------------------------------------------------
__global__ __launch_bounds__(1024) void exscan_kernel(
    const float* __restrict__ deg, int* __restrict__ off, int n) {
  __shared__ int warpsum[32];
  const int lane = threadIdx.x & 31;
  const int w    = threadIdx.x >> 5;
  int carry = 0;
  for (int base = 0; base < n; base += 1024) {
    const int i = base + (int)threadIdx.x;
    const int v = (i < n) ? (int)deg[i] : 0;
    // inclusive scan within wave
    int s = v;
#pragma unroll
    for (int o = 1; o < 32; o <<= 1) {
      int t = __shfl_up(s, o, 32);
      if (lane >= o) s += t;
    }
    if (lane == 31) warpsum[w] = s;
    __syncthreads();
    if (w == 0) {
      int ws_ = warpsum[lane];
#pragma unroll
      for (int o = 1; o < 32; o <<= 1) {
        int t = __shfl_up(ws_, o, 32);
        if (lane >= o) ws_ += t;
      }
      warpsum[lane] = ws_;
    }
    __syncthreads();
    const int incl = s + (w ? warpsum[w - 1] : 0) + carry;
    if (i < n) off[i] = incl - v;           // exclusive
    const int total = warpsum[31];
    __syncthreads();                        // protect warpsum before next chunk
    carry += total;
  }
  if (threadIdx.x == 0) off[n] = carry;
}

// ---------------------------------------------------------------------------
// CSR fill: csr_src[off[dst[e]] + cursor[dst[e]]++] = src[e]
// (one int atomic per edge; fp-order within a row is non-deterministic just
//  like any atomic accumulation, but the work is identical every call)
// ---------------------------------------------------------------------------
__global__ void csr_fill_kernel(const int* __restrict__ src, const int* __restrict__ dst,
                                const int* __restrict__ off, int* __restrict__ cursor,
                                int* __restrict__ csr_src, int n) {
  int i = blockIdx.x * blockDim.x + threadIdx.x;
  if (i < n) {
    const int d = dst[i];
    const int pos = atomicAdd(&cursor[d], 1);
    csr_src[off[d] + pos] = src[i];
  }
}

// ---------------------------------------------------------------------------
// Gather-based graph conv: out[r] (= or +=) norm_d[r] * sum_{s in N(r)} x[s]*norm_s[s]
// One wave per destination row, float4/lane register accumulation, single
// store per row -> pure gather traffic, zero fp atomics.
// Edge indices + source norms are batch-loaded (1/lane, 32 per wave) and shfl-broadcast.
// ---------------------------------------------------------------------------
template <int ACCUM>
__global__ __launch_bounds__(256) void gather_kernel(
    const float* __restrict__ x, const int* __restrict__ csr_src,
    const int* __restrict__ off, const float* __restrict__ norm_s,
    const float* __restrict__ norm_d, float* __restrict__ out, int rows) {
  const int lane = threadIdx.x & 31;
  const int r = (blockIdx.x * blockDim.x + threadIdx.x) >> 5;
  if (r >= rows) return;
  const int e0 = off[r];
  const int e1 = off[r + 1];
  float4 acc = {0.f, 0.f, 0.f, 0.f};
  for (int e = e0; e < e1; e += 32) {
    const int cnt = min(32, e1 - e);
    int myidx = 0;
    float myw = 0.f;
    if (e + lane < e1) {
      myidx = csr_src[e + lane];
      myw   = norm_s[myidx];
    }
    for (int j = 0; j < cnt; ++j) {
      const int   s = __shfl(myidx, j, 32);
      const float w = __shfl(myw,  j, 32);
      const float4 v = ((const float4*)(x + (size_t)s * HID))[lane];
      acc.x += v.x * w; acc.y += v.y * w; acc.z += v.z * w; acc.w += v.w * w;
    }
  }
  const float nd = norm_d[r];
  float4* op = (float4*)(out + (size_t)r * HID) + lane;
  if (ACCUM) {
    float4 p = *op;
    p.x += acc.x * nd; p.y += acc.y * nd; p.z += acc.z * nd; p.w += acc.w * nd;
    *op = p;
  } else {
    float4 o = { acc.x * nd, acc.y * nd, acc.z * nd, acc.w * nd };
    *op = o;
  }
}

// ---------------------------------------------------------------------------
// In-place layernorm: one wave (32 lanes x float4) per 128-wide row.
// ---------------------------------------------------------------------------
__global__ __launch_bounds__(256) void layernorm_kernel(
    float* __restrict__ x, const float* __restrict__ g, const float* __restrict__ b,
    int rows) {
  const int lane = threadIdx.x & 31;
  const int wid  = (blockIdx.x * blockDim.x + threadIdx.x) >> 5;
  if (wid >= rows) return;
  float* row = x + (size_t)wid * HID;
  float4 v = ((const float4*)row)[lane];
  float s = v.x + v.y + v.z + v.w;
#pragma unroll
  for (int m = 16; m >= 1; m >>= 1) s += __shfl_xor(s, m, 32);
  const float mu = s * (1.0f / HID);
  float4 dv = { v.x - mu, v.y - mu, v.z - mu, v.w - mu };
  float q = dv.x * dv.x + dv.y * dv.y + dv.z * dv.z + dv.w * dv.w;
#pragma unroll
  for (int m = 16; m >= 1; m >>= 1) q += __shfl_xor(q, m, 32);
  const float inv = rsqrtf(q * (1.0f / HID) + 1e-5f);
  const float4 gv = ((const float4*)g)[lane];
  const float4 bv = ((const float4*)b)[lane];
  float4 o = { dv.x * inv * gv.x + bv.x, dv.y * inv * gv.y + bv.y,
               dv.z * inv * gv.z + bv.z, dv.w * inv * gv.w + bv.w };
  ((float4*)row)[lane] = o;
}

// ---------------------------------------------------------------------------
// Author head: out[r] = log_softmax(relu_h[r] @ W2[128,40] + b2), one block/row.
// ---------------------------------------------------------------------------
__global__ __launch_bounds__(128) void author_head(
    const float* __restrict__ h, const float* __restrict__ W2,
    const float* __restrict__ b2, float* __restrict__ out) {
  __shared__ float sh[HID];
  __shared__ float slog[OUT];
  __shared__ float sred[2];
  const int r = blockIdx.x;
  const int t = threadIdx.x;
  sh[t] = h[(size_t)r * HID + t];
  __syncthreads();
  if (t < OUT) {
    float acc = b2[t];
    for (int k = 0; k < HID; ++k) acc += sh[k] * W2[k * OUT + t];
    slog[t] = acc;
  }
  __syncthreads();
  if (t == 0) {
    float m = slog[0];
    for (int j = 1; j < OUT; ++j) m = fmaxf(m, slog[j]);
    float s = 0.0f;
    for (int j = 0; j < OUT; ++j) s += expf(slog[j] - m);
    sred[0] = m;
    sred[1] = logf(s);
  }
  __syncthreads();
  if (t < OUT) out[(size_t)r * OUT + t] = slog[t] - sred[0] - sred[1];
}

// ---------------------------------------------------------------------------
extern "C" void kernel_launch(void* const* d_in, const int* in_sizes, int n_in,
                              void* d_out, int out_size, void* d_ws, size_t ws_size,
                              hipStream_t stream) {
  const float* x_author   = (const float*)d_in[0];
  const float* x_paper    = (const float*)d_in[1];
  const int*   writes_src = (const int*)d_in[2];
  const int*   writes_dst = (const int*)d_in[3];
  const int*   wb_src     = (const int*)d_in[4];
  const int*   wb_dst     = (const int*)d_in[5];
  const int*   cites_src  = (const int*)d_in[6];
  const int*   cites_dst  = (const int*)d_in[7];
  const float* We_author  = (const float*)d_in[8];
  const float* be_author  = (const float*)d_in[9];
  const float* We_paper   = (const float*)d_in[10];
  const float* be_paper   = (const float*)d_in[11];
  const float* ln_g_a     = (const float*)d_in[12];
  const float* ln_b_a     = (const float*)d_in[13];
  const float* ln_g_p     = (const float*)d_in[14];
  const float* ln_b_p     = (const float*)d_in[15];
  const float* W1_a       = (const float*)d_in[16];
  const float* b1_a       = (const float*)d_in[17];
  const float* W2_a       = (const float*)d_in[18];
  const float* b2_a       = (const float*)d_in[19];
  const float* W1_p       = (const float*)d_in[20];
  const float* b1_p       = (const float*)d_in[21];
  const float* W2_p       = (const float*)d_in[22];
  const float* b2_p       = (const float*)d_in[23];

  const int EW  = in_sizes[2];
  const int EWB = in_sizes[4];
  const int EC  = in_sizes[6];

  // Workspace layout: ha0|hp0|ha1|hp1|norms|CSR structures (~103 MB)
  float* ws = (float*)d_ws;
  const size_t szA = (size_t)NA * HID;
  const size_t szP = (size_t)NP * HID;
  float* ha0 = ws;
  float* hp0 = ha0 + szA;
  float* ha1 = hp0 + szP;
  float* hp1 = ha1 + szA;
  float* norms = hp1 + szP;
  float* nW_s = norms;            // [NA] writes src norm
  float* nW_d = nW_s + NA;        // [NP] writes dst norm
  float* nB_s = nW_d + NP;        // [NP] written_by src norm
  float* nB_d = nB_s + NP;        // [NA] written_by dst norm
  float* nC_s = nB_d + NA;        // [NP] cites src norm
  float* nC_d = nC_s + NP;        // [NP] cites dst norm
  const int normsTotal = 2 * NA + 4 * NP;

  int* offW = (int*)(norms + normsTotal); // [NP+1] writes->paper row offsets
  int* offB = offW + (NP + 1);            // [NA+1] written_by->author
  int* offC = offB + (NA + 1);            // [NP+1] cites->paper
  int* curW = offC + (NP + 1);            // [NP] fill cursors
  int* curB = curW + NP;                  // [NA]
  int* curC = curB + NA;                  // [NP]
  int* csrW = curC + NP;                  // [EW]  source ids grouped by dst
  int* csrB = csrW + EW;                  // [EWB]
  int* csrC = csrB + EWB;                 // [EC]
  const int cursTotal = 2 * NP + NA;

  // Zero degree arrays + CSR fill cursors.
  hipMemsetAsync(norms, 0, (size_t)normsTotal * sizeof(float), stream);
  hipMemsetAsync(curW, 0, (size_t)cursTotal * sizeof(int), stream);

  // Degrees (raw counts), then CSR row offsets from dst-degrees, then norms.
  degree_kernel<<<(EW  + 255) / 256, 256, 0, stream>>>(writes_src, writes_dst, nW_s, nW_d, EW);
  degree_kernel<<<(EWB + 255) / 256, 256, 0, stream>>>(wb_src,     wb_dst,     nB_s, nB_d, EWB);
  degree_kernel<<<(EC  + 255) / 256, 256, 0, stream>>>(cites_src,  cites_dst,  nC_s, nC_d, EC);
  exscan_kernel<<<1, 1024, 0, stream>>>(nW_d, offW, NP);
  exscan_kernel<<<1, 1024, 0, stream>>>(nB_d, offB, NA);
  exscan_kernel<<<1, 1024, 0, stream>>>(nC_d, offC, NP);
  rsqrt_clamp_kernel<<<(normsTotal + 255) / 256, 256, 0, stream>>>(norms, normsTotal);

  // CSR fill (built once, reused by both layers).
  csr_fill_kernel<<<(EW  + 255) / 256, 256, 0, stream>>>(writes_src, writes_dst, offW, curW, csrW, EW);
  csr_fill_kernel<<<(EWB + 255) / 256, 256, 0, stream>>>(wb_src,     wb_dst,     offB, curB, csrB, EWB);
  csr_fill_kernel<<<(EC  + 255) / 256, 256, 0, stream>>>(cites_src,  cites_dst,  offC, curC, csrC, EC);

  // Input embeddings (fp32 WMMA GEMMs).
  {
    const int tiles = (NA / 16) * (HID / 16);
    gemm_wmma<0><<<(tiles + 7) / 8, 256, 0, stream>>>(x_author, We_author, be_author, ha0, NA, HID, FA);
  }
  {
    const int tiles = (NP / 16) * (HID / 16);
    gemm_wmma<0><<<(tiles + 7) / 8, 256, 0, stream>>>(x_paper, We_paper, be_paper, hp0, NP, HID, FP);
  }

  const int GA = (NA + 7) / 8;  // author-rows gather blocks (8 waves/block)
  const int GP = (NP + 7) / 8;  // paper-rows gather blocks

  // Layer 1: ha1 = gconv(hp0, wb); hp1 = gconv(ha0, writes) + gconv(hp0, cites)
  gather_kernel<0><<<GA, 256, 0, stream>>>(hp0, csrB, offB, nB_s, nB_d, ha1, NA);
  gather_kernel<0><<<GP, 256, 0, stream>>>(ha0, csrW, offW, nW_s, nW_d, hp1, NP);
  gather_kernel<1><<<GP, 256, 0, stream>>>(hp0, csrC, offC, nC_s, nC_d, hp1, NP);

  // Layer 2: ha0 = gconv(hp1, wb); hp0 = gconv(ha1, writes) + gconv(hp1, cites)
  gather_kernel<0><<<GA, 256, 0, stream>>>(hp1, csrB, offB, nB_s, nB_d, ha0, NA);
  gather_kernel<0><<<GP, 256, 0, stream>>>(ha1, csrW, offW, nW_s, nW_d, hp0, NP);
  gather_kernel<1><<<GP, 256, 0, stream>>>(hp1, csrC, offC, nC_s, nC_d, hp0, NP);

  // LayerNorm in place.
  layernorm_kernel<<<(NA + 7) / 8, 256, 0, stream>>>(ha0, ln_g_a, ln_b_a, NA);
  layernorm_kernel<<<(NP + 7) / 8, 256, 0, stream>>>(hp0, ln_g_p, ln_b_p, NP);

  // Author head: relu(ha0@W1+b1) -> ha1; then 128x40 + log_softmax fused.
  {
    const int tiles = (NA / 16) * (HID / 16);
    gemm_wmma<1><<<(tiles + 7) / 8, 256, 0, stream>>>(ha0, W1_a, b1_a, ha1, NA, HID, HID);
    author_head<<<NA, 128, 0, stream>>>(ha1, W2_a, b2_a, (float*)d_out);
  }
  // Paper head: relu(hp0@W1+b1) -> hp1; hp1@W2+b2 -> out.
  {
    const int tiles = (NP / 16) * (HID / 16);
    gemm_wmma<1><<<(tiles + 7) / 8, 256, 0, stream>>>(hp0, W1_p, b1_p, hp1, NP, HID, HID);
    gemm_wmma<0><<<(tiles + 7) / 8, 256, 0, stream>>>(hp1, W2_p, b2_p,
                                                      (float*)d_out + (size_t)NA * OUT,
                                                      NP, HID, HID);
  }
}